// GreedySearch_58213986730356
// MI455X (gfx1250) — compile-verified
//
#include <hip/hip_runtime.h>
#include <hip/hip_bf16.h>

// ---------------------------------------------------------------------------
// GreedySearch on MI455X (gfx1250).
//
// x / lens are dead inputs (dataflow-proven). The computation reduces to a
// 16-step serial loop of two small GEMMs + per-batch argmin over C=512.
// Single persistent workgroup (32 wave32s on one WGP):
//   - pls (32x2048 f32 = 256KB) resident in CDNA5's 320KB LDS
//   - both GEMMs on f32 WMMA (v_wmma_f32_16x16x4_f32), full f32 precision
//   - W pre-transposed into d_ws so B-fragments load as contiguous b64
//   - step-B K-loop shares each labels B-fragment across two M-tile WMMAs
//   - argmin via monotone-key u64 packing + ds_min_u64 (ties -> smallest c)
// ---------------------------------------------------------------------------

typedef __attribute__((ext_vector_type(2))) float v2f;
typedef __attribute__((ext_vector_type(8))) float v8f;

#define NB    32              // batch B
#define TL    16              // T_L
#define NJ    128             // J
#define NC    512             // C
#define KFLAT (TL * NJ)       // 2048

// Dynamic-LDS byte offsets (total 295,808 B <= 320KB WGP LDS)
#define OFF_PLS   0                          // 32*2048*4   = 262144
#define OFF_L2    (NB * KFLAT * 4)           // 512*16*4    =  32768
#define OFF_P0    (OFF_L2 + NC * TL * 4)     // 128*4       =    512
#define OFF_AM    (OFF_P0 + NJ * 4)          // 32*8        =    256 (8-aligned)
#define OFF_SOFAR (OFF_AM + NB * 8)          // 32*4        =    128
#define SMEM_BYTES (OFF_SOFAR + NB * 4)      // 295808

// Monotone float->unsigned key packed with index: u64-min == argmin (ties: min c).
__device__ inline unsigned long long pack_key(float s, int c) {
    unsigned b = __float_as_uint(s);
    unsigned key = (b & 0x80000000u) ? ~b : (b | 0x80000000u);
    return ((unsigned long long)key << 32) | (unsigned)c;
}

// One-time 128x128 transpose: Wt[n][k] = W[k][n]
__global__ void transposeW_kernel(const float* __restrict__ W, float* __restrict__ Wt) {
    int i = threadIdx.x + blockIdx.x * blockDim.x;   // 0..16383
    int n = i >> 7, k = i & 127;
    Wt[n * NJ + k] = W[k * NJ + n];
}

// USE_WT: Wmat layout is Wt[n][k] (transposed, contiguous fragment loads).
// else : Wmat layout is W[k][n]  (strided fallback if d_ws is too small).
template <bool USE_WT>
__global__ __launch_bounds__(1024, 1)
void GreedySearch_58213986730356_kernel(const float* __restrict__ sos,
                                        const float* __restrict__ labels, // [C,TL,J]
                                        const float* __restrict__ Wmat,
                                        float* __restrict__ out)          // [32 + 32*16*128]
{
    extern __shared__ char smem[];
    float*              sh_pls   = (float*)(smem + OFF_PLS);   // [NB][KFLAT]
    float*              sh_l2    = (float*)(smem + OFF_L2);    // [NC][TL] inclusive prefix
    float*              sh_p0    = (float*)(smem + OFF_P0);    // [NJ]
    unsigned long long* sh_am    = (unsigned long long*)(smem + OFF_AM);  // [NB]
    int*                sh_sofar = (int*)(smem + OFF_SOFAR);   // [NB]

    const int tid  = threadIdx.x;
    const int nthr = blockDim.x;       // 1024
    const int wave = tid >> 5;         // 0..31
    const int lane = tid & 31;
    const int half = lane >> 4;        // WMMA half-wave select
    const int l16  = lane & 15;

    const v8f vzero = {0.f, 0.f, 0.f, 0.f, 0.f, 0.f, 0.f, 0.f};

    // ---------------- Phase 0: l2 norms, prefix sums, p0 = sos @ W ----------
    for (int task = tid; task < NC * TL; task += nthr) {
        const int c = task >> 4, tt = task & 15;
        const float* p = labels + (size_t)c * KFLAT + tt * NJ;
        float s = 0.f;
        #pragma unroll 8
        for (int j = 0; j < NJ; ++j) s += p[j] * p[j];
        sh_l2[c * TL + tt] = s;
    }
    __syncthreads();
    for (int c = tid; c < NC; c += nthr) {       // inclusive prefix over tt
        float run = 0.f;
        #pragma unroll
        for (int tt = 0; tt < TL; ++tt) { run += sh_l2[c * TL + tt]; sh_l2[c * TL + tt] = run; }
    }
    if (tid < NJ) {                              // p0[k] = sum_j sos[j]*W[j][k]
        float s = 0.f;
        #pragma unroll 8
        for (int j = 0; j < NJ; ++j)
            s += sos[j] * (USE_WT ? Wmat[tid * NJ + j] : Wmat[j * NJ + tid]);
        sh_p0[tid] = s;
    }
    if (tid < NB) sh_am[tid] = ~0ULL;
    __syncthreads();

    // Initial argmin: score0[c] = ||labels[c,0]||^2 - 2 * p0 . labels[c,0]
    for (int c = tid; c < NC; c += nthr) {
        const float* lc = labels + (size_t)c * KFLAT;
        float dot = 0.f;
        #pragma unroll 8
        for (int j = 0; j < NJ; ++j) dot += sh_p0[j] * lc[j];
        atomicMin(&sh_am[0], pack_key(sh_l2[c * TL] - 2.0f * dot, c));
    }
    __syncthreads();
    if (tid < NB) sh_sofar[tid] = (int)(unsigned)(sh_am[0] & 0xFFFFFFFFull);
    __syncthreads();
    if (tid < NB) sh_am[tid] = ~0ULL;
    __syncthreads();

    // ---------------- Main serial loop: t = 1..16 ----------------------------
    for (int t = 1; t <= TL; ++t) {
        // Step A: pls[b] = labels[sofar[b]] @ W  (wave w owns batch b = w).
        // K-outer / 4-wide n-tile groups: A-fragment loaded once per k-step.
        {
            const int b = wave;
            const float* lab = labels + (size_t)sh_sofar[b] * KFLAT;  // [16 x 128]
            #pragma unroll
            for (int ng = 0; ng < 2; ++ng) {                          // n-tile groups of 4
                v8f acc[4];
                acc[0] = vzero; acc[1] = vzero; acc[2] = vzero; acc[3] = vzero;
                #pragma unroll 4
                for (int kk = 0; kk < NJ / 4; ++kk) {                 // K = 128
                    const int k = 4 * kk + 2 * half;
                    v2f a;
                    a.x = lab[l16 * NJ + k];
                    a.y = lab[l16 * NJ + k + 1];
                    #pragma unroll
                    for (int q = 0; q < 4; ++q) {
                        const int n0 = ng * 4 + q;
                        v2f bm;
                        if (USE_WT) {
                            const float* wt = Wmat + (size_t)(n0 * 16 + l16) * NJ;
                            bm.x = wt[k];                   // contiguous -> b64
                            bm.y = wt[k + 1];
                        } else {
                            bm.x = Wmat[(k    ) * NJ + n0 * 16 + l16];
                            bm.y = Wmat[(k + 1) * NJ + n0 * 16 + l16];
                        }
                        acc[q] = __builtin_amdgcn_wmma_f32_16x16x4_f32(
                            false, a, false, bm, (short)0, acc[q], false, false);
                    }
                }
                #pragma unroll
                for (int q = 0; q < 4; ++q) {
                    const int n0 = ng * 4 + q;
                    #pragma unroll
                    for (int r = 0; r < 8; ++r)
                        sh_pls[b * KFLAT + (r + 8 * half) * NJ + n0 * 16 + l16] = acc[q][r];
                }
            }
        }
        __syncthreads();

        // Step B: cross GEMM  M=32(b) x N=512(c), K = t*128 (mask via K-range).
        // One K-loop, two accumulators: each labels B-fragment feeds 2 WMMAs.
        {
            const int Ksteps = t * (NJ / 4);
            const int c = wave * 16 + l16;                       // wave owns N-tile = wave
            const float* lc  = labels + (size_t)c * KFLAT;       // B[k][c] = labels[c][k]
            const float  l2p = sh_l2[c * TL + (t - 1)];          // sum_{tt<t} ||labels[c,tt]||^2
            const float* pr0 = sh_pls + (size_t)(l16     ) * KFLAT;   // M-tile 0 rows
            const float* pr1 = sh_pls + (size_t)(l16 + 16) * KFLAT;   // M-tile 1 rows
            v8f acc0 = vzero, acc1 = vzero;
            #pragma unroll 4
            for (int kk = 0; kk < Ksteps; ++kk) {
                const int k = 4 * kk + 2 * half;
                v2f bm, a0, a1;
                bm.x = lc[k];   bm.y = lc[k + 1];                // global (L2-hot), b64
                a0.x = pr0[k];  a0.y = pr0[k + 1];               // LDS, b64
                a1.x = pr1[k];  a1.y = pr1[k + 1];               // LDS, b64
                acc0 = __builtin_amdgcn_wmma_f32_16x16x4_f32(
                    false, a0, false, bm, (short)0, acc0, false, false);
                acc1 = __builtin_amdgcn_wmma_f32_16x16x4_f32(
                    false, a1, false, bm, (short)0, acc1, false, false);
            }
            // score[b,c] = l2prefix - 2*cross ; argmin-reduce over the 16 c-lanes
            #pragma unroll
            for (int mt = 0; mt < 2; ++mt) {
                const v8f& acc = mt ? acc1 : acc0;
                #pragma unroll
                for (int r = 0; r < 8; ++r) {
                    unsigned long long p = pack_key(l2p - 2.0f * acc[r], c);
                    #pragma unroll
                    for (int off = 8; off > 0; off >>= 1) {
                        unsigned long long o = __shfl_xor(p, off, 32);
                        p = (o < p) ? o : p;
                    }
                    if (l16 == 0)
                        atomicMin(&sh_am[mt * 16 + r + 8 * half], p);
                }
            }
        }
        __syncthreads();
        if (tid < NB) {
            sh_sofar[tid] = (int)(unsigned)(sh_am[tid] & 0xFFFFFFFFull);
            sh_am[tid] = ~0ULL;                                  // reset for next iter
        }
        __syncthreads();
    }

    // ---------------- Output: [sofar (as float) | pls] -----------------------
    if (tid < NB) out[tid] = (float)sh_sofar[tid];
    for (int i = tid; i < NB * KFLAT; i += nthr) out[NB + i] = sh_pls[i];
}

extern "C" void kernel_launch(void* const* d_in, const int* in_sizes, int n_in,
                              void* d_out, int out_size, void* d_ws, size_t ws_size,
                              hipStream_t stream) {
    // setup_inputs order: x[0], lens[1], sos[2], label_seqs[3], W[4]
    // x and lens are dead inputs (proven by dataflow) — not read.
    const float* sos    = (const float*)d_in[2];
    const float* labels = (const float*)d_in[3];
    const float* W      = (const float*)d_in[4];
    float* out = (float*)d_out;
    (void)in_sizes; (void)n_in; (void)out_size;

    if (ws_size >= (size_t)(NJ * NJ * sizeof(float))) {
        float* Wt = (float*)d_ws;
        transposeW_kernel<<<16, 1024, 0, stream>>>(W, Wt);
        GreedySearch_58213986730356_kernel<true><<<1, 1024, SMEM_BYTES, stream>>>(
            sos, labels, Wt, out);
    } else {
        GreedySearch_58213986730356_kernel<false><<<1, 1024, SMEM_BYTES, stream>>>(
            sos, labels, W, out);
    }
}